// GraphSAGE_50766513439525
// MI455X (gfx1250) — compile-verified
//
#include <hip/hip_runtime.h>

typedef float v2f __attribute__((ext_vector_type(2)));
typedef float v8f __attribute__((ext_vector_type(8)));

#define DK 64   // input feature dim of both layers (D_IN == D_H == 64)

// ---------------------------------------------------------------------------
// Scatter phase: agg[dst] += feat[src] (and optionally deg[dst] += 1).
// One thread per (edge, 16-byte chunk): float4 load + 4 f32 global atomics.
// 16 consecutive lanes cover one edge's 64-float row -> coalesced reads.
// Atomics land in the 192MB L2 (agg buffer is only 25.6MB -> L2 resident).
// ---------------------------------------------------------------------------
__global__ void __launch_bounds__(256)
sage_scatter(const int* __restrict__ src, const int* __restrict__ dst,
             const float* __restrict__ feat, float* __restrict__ agg,
             float* __restrict__ deg, int nedges, int add_deg)
{
    int tid = blockIdx.x * blockDim.x + threadIdx.x;
    int e = tid >> 4;
    if (e >= nedges) return;
    int c = (tid & 15) << 2;              // float offset within row: 0..60
    int s = src[e];
    int d = dst[e];
    float4 v = *(const float4*)(feat + (size_t)s * DK + c);
    float* ap = agg + (size_t)d * DK + c;
    atomicAdd(ap + 0, v.x);
    atomicAdd(ap + 1, v.y);
    atomicAdd(ap + 2, v.z);
    atomicAdd(ap + 3, v.w);
    if (add_deg && (tid & 15) == 0) atomicAdd(deg + d, 1.0f);
}

// ---------------------------------------------------------------------------
// Fused SAGE linear: out = act( mean(agg,deg) @ Wl^T + self @ Wr^T + bias )
// One wave32 handles one 16-row node tile using V_WMMA_F32_16X16X4_F32.
//   A (16x4 f32): lane L holds row M=L%16, K-pair {2*(L/16), 2*(L/16)+1}
//   B (4x16 f32): lane L holds col N=L%16, same K-pair selection
//   C (16x16 f32): VGPR r, lane L -> M = r + 8*(L/16), N = L%16
// The 1/max(deg,1) mean scaling is fused into the A load; bias (+ReLU) fused
// into the store. All branches are wave-uniform => EXEC all-1s for WMMA.
// ---------------------------------------------------------------------------
template<int DOUT, bool RELU>
__global__ void __launch_bounds__(256)
sage_gemm(const float* __restrict__ agg, const float* __restrict__ deg,
          const float* __restrict__ self, const float* __restrict__ Wl,
          const float* __restrict__ Wr, const float* __restrict__ bias,
          float* __restrict__ out, int nnodes)
{
    const int lane  = threadIdx.x & 31;
    const int wave  = threadIdx.x >> 5;
    const int tile  = blockIdx.x * (blockDim.x >> 5) + wave;
    const int mbase = tile << 4;
    if (mbase >= nnodes) return;          // wave-uniform exit

    const int   row   = mbase + (lane & 15);
    const int   khalf = (lane >> 4) << 1;          // 0 or 2
    const float inv   = 1.0f / fmaxf(deg[row], 1.0f);
    const float* arow = agg  + (size_t)row * DK;
    const float* srow = self + (size_t)row * DK;

    #pragma unroll
    for (int nb = 0; nb < DOUT; nb += 16) {
        const int    ncol = nb + (lane & 15);
        const float* wl   = Wl + (size_t)ncol * DK;   // Wl is [DOUT, DK] row-major
        const float* wr   = Wr + (size_t)ncol * DK;

        v8f c = {};

        // neighbor-mean branch: A = mean tile, B = Wl^T
        #pragma unroll
        for (int kb = 0; kb < DK; kb += 4) {
            const int k = kb + khalf;
            v2f a, b;
            a.x = arow[k]     * inv;
            a.y = arow[k + 1] * inv;
            b.x = wl[k];
            b.y = wl[k + 1];
            c = __builtin_amdgcn_wmma_f32_16x16x4_f32(
                    false, a, false, b, (short)0, c, false, false);
        }
        // root branch: A = self features, B = Wr^T
        #pragma unroll
        for (int kb = 0; kb < DK; kb += 4) {
            const int k = kb + khalf;
            v2f a, b;
            a.x = srow[k];
            a.y = srow[k + 1];
            b.x = wr[k];
            b.y = wr[k + 1];
            c = __builtin_amdgcn_wmma_f32_16x16x4_f32(
                    false, a, false, b, (short)0, c, false, false);
        }

        const float bv = bias[ncol];
        float* obase = out + ((size_t)mbase + ((lane >> 4) << 3)) * DOUT + ncol;
        #pragma unroll
        for (int r = 0; r < 8; ++r) {
            float v = c[r] + bv;
            if (RELU) v = fmaxf(v, 0.0f);
            obase[(size_t)r * DOUT] = v;
        }
    }
}

// ---------------------------------------------------------------------------
// Inputs (setup_inputs order): x[N*64], edge_index[2*E], W1_l[64*64],
// W1_r[64*64], b1[64], W2_l[32*64], W2_r[32*64], b2[32]. Output: [N*32] f32.
// Workspace: deg (N f32) | agg (N*64 f32) | h (N*64 f32)  ~= 51.6 MB.
// ---------------------------------------------------------------------------
extern "C" void kernel_launch(void* const* d_in, const int* in_sizes, int n_in,
                              void* d_out, int out_size, void* d_ws, size_t ws_size,
                              hipStream_t stream)
{
    const float* x   = (const float*)d_in[0];
    const int*   ei  = (const int*)d_in[1];
    const float* W1l = (const float*)d_in[2];
    const float* W1r = (const float*)d_in[3];
    const float* b1  = (const float*)d_in[4];
    const float* W2l = (const float*)d_in[5];
    const float* W2r = (const float*)d_in[6];
    const float* b2  = (const float*)d_in[7];
    float*       out = (float*)d_out;

    const int nN = in_sizes[0] / DK;
    const int nE = in_sizes[1] / 2;
    const int* src = ei;
    const int* dst = ei + nE;

    float* deg = (float*)d_ws;
    float* agg = deg + nN;
    float* h   = agg + (size_t)nN * DK;

    const int sblocks = (nE * 16 + 255) / 256;
    const int tiles   = (nN + 15) / 16;
    const int gblocks = (tiles + 7) / 8;          // 8 waves / 256-thread block

    // ---- layer 1 ----
    hipMemsetAsync(deg, 0, (size_t)nN * (DK + 1) * sizeof(float), stream);
    sage_scatter<<<sblocks, 256, 0, stream>>>(src, dst, x, agg, deg, nE, 1);
    sage_gemm<64, true><<<gblocks, 256, 0, stream>>>(agg, deg, x, W1l, W1r, b1, h, nN);

    // ---- layer 2 (degree unchanged; reuse agg buffer) ----
    hipMemsetAsync(agg, 0, (size_t)nN * DK * sizeof(float), stream);
    sage_scatter<<<sblocks, 256, 0, stream>>>(src, dst, h, agg, deg, nE, 0);
    sage_gemm<32, false><<<gblocks, 256, 0, stream>>>(agg, deg, h, W2l, W2r, b2, out, nN);
}